// GraphQNN__gen282_65481071402774
// MI455X (gfx1250) — compile-verified
//
#include <hip/hip_runtime.h>
#include <hip/hip_bf16.h>

// ---------------------------------------------------------------------------
// 16-qubit statevector QNN on MI455X (gfx1250, wave32).
// State: (256, 2^16) complex64, interleaved float2, 128 MB.
// Strategy:
//   * encoding RYs on |0..0> => product state written analytically (1 pass)
//   * each layer: 4 passes, one fused 16x16 complex unitary per 4-qubit group
//     (RZ*RY*RX per qubit + in-group CNOTs folded into the matrix; cross-group
//     CNOT folded into the NEXT pass's row addressing: k ^= 8 when the control
//     bit -- part of the outer index -- is 1)
//   * each 16x16 complex matmul = 4 real f32 matmuls = 16 V_WMMA_F32_16X16X4_F32
// 13 total passes over 128 MB => ~3.3 GB @ 23.3 TB/s => memory bound ~145us.
// ---------------------------------------------------------------------------

typedef float v2f __attribute__((ext_vector_type(2)));
typedef float v8f __attribute__((ext_vector_type(8)));

// Assert the builtin only in the DEVICE pass; __has_builtin is 0 on the host
// pass (x86), which is expected and harmless.
#if defined(__HIP_DEVICE_COMPILE__)
#if !__has_builtin(__builtin_amdgcn_wmma_f32_16x16x4_f32)
#error "device pass: missing __builtin_amdgcn_wmma_f32_16x16x4_f32 for gfx1250"
#endif
#endif

__device__ __forceinline__ v8f wmma4(v2f a, v2f b, v8f c) {
  // 8 args: (neg_a, A, neg_b, B, c_mod, C, reuse_a, reuse_b); f32 has no A/B neg.
  return __builtin_amdgcn_wmma_f32_16x16x4_f32(false, a, false, b, (short)0, c,
                                               false, false);
}

// ------------------------- small complex helpers ---------------------------
struct cplx { float r, i; };
__device__ __forceinline__ cplx cmul(cplx a, cplx b) {
  return {a.r * b.r - a.i * b.i, a.r * b.i + a.i * b.r};
}
__device__ __forceinline__ cplx cadd(cplx a, cplx b) { return {a.r + b.r, a.i + b.i}; }

__device__ __forceinline__ void mul2(const cplx A[2][2], const cplx B[2][2],
                                     cplx C[2][2]) {
#pragma unroll
  for (int i = 0; i < 2; ++i)
#pragma unroll
    for (int j = 0; j < 2; ++j)
      C[i][j] = cadd(cmul(A[i][0], B[0][j]), cmul(A[i][1], B[1][j]));
}

// ---------------------------------------------------------------------------
// Kernel 1: per-sample nibble amplitude tables for the RY encoding layer.
// tab[b][g][v] = prod_{j=0..3} (bit(3-j) of v ? sin : cos)(x[b][4g+j]/2)
// ---------------------------------------------------------------------------
__global__ void qnn_tables(const float* __restrict__ x, float* __restrict__ tab) {
  const int idx = blockIdx.x * 256 + threadIdx.x;  // 256*4*16 = 16384 entries
  if (idx >= 256 * 64) return;
  const int b = idx >> 6;
  const int g = (idx >> 4) & 3;
  const int v = idx & 15;
  float p = 1.f;
#pragma unroll
  for (int j = 0; j < 4; ++j) {
    const float t = 0.5f * x[b * 16 + 4 * g + j];
    const int bit = (v >> (3 - j)) & 1;
    p *= bit ? sinf(t) : cosf(t);
  }
  tab[idx] = p;
}

// ---------------------------------------------------------------------------
// Kernel 2: build the 12 fused 16x16 complex unitaries (3 layers x 4 groups).
//   V[m][k] = prod_j (RZ*RY*RX)_{qubit 4g+j}[bit(3-j) of m][bit(3-j) of k]
//   U[perm(m)][k] = V[m][k]   (in-group CNOT chain as row scatter)
//   perm: b2^=b3; b1^=b2; b0^=b1  (sequential)
// Params: layer base = 48*layer, 3 per qubit (RX, RY, RZ). CNOTs use none.
// ---------------------------------------------------------------------------
__global__ void qnn_build(const float* __restrict__ params,
                          float2* __restrict__ Uall) {
  const int pass = blockIdx.x;  // 0..11
  const int layer = pass >> 2;
  const int g = pass & 3;
  const int m = threadIdx.x >> 4;
  const int k = threadIdx.x & 15;

  cplx prod = {1.f, 0.f};
#pragma unroll
  for (int j = 0; j < 4; ++j) {
    const int q = 4 * g + j;
    const float* p = params + layer * 48 + q * 3;
    const float hx = 0.5f * p[0], hy = 0.5f * p[1], hz = 0.5f * p[2];
    const float cx = cosf(hx), sx = sinf(hx);
    const float cy = cosf(hy), sy = sinf(hy);
    const float cz = cosf(hz), sz = sinf(hz);
    const cplx RX[2][2] = {{{cx, 0.f}, {0.f, -sx}}, {{0.f, -sx}, {cx, 0.f}}};
    const cplx RY[2][2] = {{{cy, 0.f}, {-sy, 0.f}}, {{sy, 0.f}, {cy, 0.f}}};
    const cplx RZ[2][2] = {{{cz, -sz}, {0.f, 0.f}}, {{0.f, 0.f}, {cz, sz}}};
    cplx T[2][2], M[2][2];
    mul2(RY, RX, T);   // RY*RX
    mul2(RZ, T, M);    // RZ*RY*RX (RX applied first)
    const int mb = (m >> (3 - j)) & 1;
    const int kb = (k >> (3 - j)) & 1;
    prod = cmul(prod, M[mb][kb]);
  }
  const int b3 = (m >> 3) & 1, b2 = (m >> 2) & 1, b1 = (m >> 1) & 1, b0 = m & 1;
  const int p2 = b2 ^ b3;
  const int p1 = b1 ^ p2;
  const int p0 = b0 ^ p1;
  const int pm = (b3 << 3) | (p2 << 2) | (p1 << 1) | p0;
  Uall[pass * 256 + pm * 16 + k] = make_float2(prod.r, prod.i);
}

// ---------------------------------------------------------------------------
// Kernel 3: write the encoded product state (real, imag = 0).
// ---------------------------------------------------------------------------
__global__ __launch_bounds__(256) void qnn_init(const float* __restrict__ tab,
                                                float2* __restrict__ st) {
  const int gid = blockIdx.x * 256 + threadIdx.x;  // 2^24 amplitudes
  const int b = gid >> 16;
  const int i = gid & 0xFFFF;
  const float* t = tab + b * 64;
  const float v = t[(i >> 12) & 15] * t[16 + ((i >> 8) & 15)] *
                  t[32 + ((i >> 4) & 15)] * t[48 + (i & 15)];
  st[gid] = make_float2(v, 0.f);
}

// ---------------------------------------------------------------------------
// Kernel 4: apply one group unitary. State viewed as (B, L=16^G, 16, R=16^(3-G));
// each wave computes one 16x16 complex tile D = U * S with 16 WMMA f32 ops.
// FOLD: cross-CNOT from previous group => read S rows as (k ^ 8) when the
// control bit (LSB of the outer index / column outer index for G==3) is 1.
// ---------------------------------------------------------------------------
template <int G, bool FOLD>
__global__ __launch_bounds__(256) void qnn_apply(const float2* __restrict__ src,
                                                 float2* __restrict__ dst,
                                                 const float2* __restrict__ Uc) {
  const int lane = threadIdx.x & 31;
  const int wave = threadIdx.x >> 5;
  const int tile = blockIdx.x * 8 + wave;  // 65536 tiles total
  const int h = lane >> 4;                 // lane half
  const int n = lane & 15;                 // column within tile

  int colBase, rs, flip;
  if constexpr (G < 3) {
    constexpr int Rsh = 4 * (3 - G);          // row stride = 16^(3-G)
    constexpr int ctSh = Rsh - 4;             // column-tiles per outer = R/16
    const int o = tile >> ctSh;               // outer index (b*L + l)
    const int ct = tile & ((1 << ctSh) - 1);
    const int base = (o << (Rsh + 4)) + (ct << 4);
    rs = 1 << Rsh;
    colBase = base + n;                       // column stride 1 (along r)
    flip = (FOLD && (o & 1)) ? 8 : 0;         // ctrl = qubit 4G-1 = LSB of l
  } else {
    const int base = tile << 8;               // rows = innermost nibble
    rs = 1;
    colBase = base + n * 16;                  // columns = 16 outer indices
    flip = (FOLD && (n & 1)) ? 8 : 0;         // ctrl per column
  }

  v2f ar[4], ai[4], an[4], br[4], bi[4];
#pragma unroll
  for (int c = 0; c < 4; ++c) {
    const int kA = 4 * c + 2 * h;             // K indices for this lane
    const int kB = kA + 1;
    const float2 u0 = Uc[n * 16 + kA];        // A row m == lane&15
    const float2 u1 = Uc[n * 16 + kB];
    ar[c].x = u0.x; ar[c].y = u1.x;
    ai[c].x = u0.y; ai[c].y = u1.y;
    an[c] = -ai[c];
    const float2 s0 = src[colBase + (kA ^ flip) * rs];
    const float2 s1 = src[colBase + (kB ^ flip) * rs];
    br[c].x = s0.x; br[c].y = s1.x;
    bi[c].x = s0.y; bi[c].y = s1.y;
  }

  v8f Dr = {};
  v8f Di = {};
#pragma unroll
  for (int c = 0; c < 4; ++c) {
    Dr = wmma4(ar[c], br[c], Dr);   // += Ur*Sr
    Dr = wmma4(an[c], bi[c], Dr);   // += (-Ui)*Si
    Di = wmma4(ar[c], bi[c], Di);   // += Ur*Si
    Di = wmma4(ai[c], br[c], Di);   // += Ui*Sr
  }

#pragma unroll
  for (int j = 0; j < 8; ++j) {
    const int m = j + 8 * h;        // D row for VGPR j, lane half h
    dst[colBase + m * rs] = make_float2(Dr[j], Di[j]);
  }
}

// ---------------------------------------------------------------------------
extern "C" void kernel_launch(void* const* d_in, const int* in_sizes, int n_in,
                              void* d_out, int out_size, void* d_ws,
                              size_t ws_size, hipStream_t stream) {
  (void)in_sizes; (void)n_in; (void)out_size; (void)ws_size;
  const float* x = (const float*)d_in[0];       // (256, 16) f32
  const float* params = (const float*)d_in[1];  // (189,)   f32

  float2* bufA = (float2*)d_out;                          // 2^24 c64 = 128 MB
  float2* bufB = (float2*)d_ws;                           // ping buffer 128 MB
  float2* Uall = (float2*)((char*)d_ws + ((size_t)1 << 27));  // 12*256 c64
  float* tab = (float*)(Uall + 12 * 256);                 // 16384 f32

  qnn_tables<<<64, 256, 0, stream>>>(x, tab);
  qnn_build<<<12, 256, 0, stream>>>(params, Uall);
  qnn_init<<<65536, 256, 0, stream>>>(tab, bufA);

  float2* src = bufA;
  float2* dst = bufB;
  for (int layer = 0; layer < 3; ++layer) {
    const float2* U = Uall + (size_t)layer * 4 * 256;
    qnn_apply<0, false><<<8192, 256, 0, stream>>>(src, dst, U + 0 * 256);
    { float2* t = src; src = dst; dst = t; }
    qnn_apply<1, true><<<8192, 256, 0, stream>>>(src, dst, U + 1 * 256);
    { float2* t = src; src = dst; dst = t; }
    qnn_apply<2, true><<<8192, 256, 0, stream>>>(src, dst, U + 2 * 256);
    { float2* t = src; src = dst; dst = t; }
    qnn_apply<3, true><<<8192, 256, 0, stream>>>(src, dst, U + 3 * 256);
    { float2* t = src; src = dst; dst = t; }
  }
  // 12 passes (even) => final state is back in bufA == d_out.
}